// OscillatoryBifurcationLayer_41540923687081
// MI455X (gfx1250) — compile-verified
//
#include <hip/hip_runtime.h>
#include <hip/hip_bf16.h>

typedef float v2f __attribute__((ext_vector_type(2)));
typedef float v4f __attribute__((ext_vector_type(4)));
typedef float v8f __attribute__((ext_vector_type(8)));
typedef unsigned int v4u __attribute__((ext_vector_type(4)));
typedef int v4i __attribute__((ext_vector_type(4)));
typedef int v8i __attribute__((ext_vector_type(8)));

#define BM 128
#define BN 128
#define BK 16
#define LDA 20    // A tile LDS stride (floats): 16 data + 4 pad -> 80B rows (16B aligned)
#define BKP 18    // transposed-B tile LDS stride (floats): even -> 8B-aligned (k,k+1) pairs

#if __has_builtin(__builtin_amdgcn_tensor_load_to_lds) && __has_builtin(__builtin_amdgcn_s_wait_tensorcnt)
#define USE_TDM 1
#else
#define USE_TDM 0
#endif

// Raw LDS byte offset of a __shared__ object (generic -> AS(3) addrspacecast).
typedef __attribute__((address_space(3))) const char lds_cchar_t;
static __device__ __forceinline__ unsigned lds_offset_of(const void* p)
{
    return (unsigned)(unsigned long long)(lds_cchar_t*)p;
}

#if USE_TDM
// Issue a TDM DMA: 128 rows x 16 f32 tile from global (row stride = K elements)
// into LDS at lds_off with a 4-DWORD pad after every 16 DWORDs (-> 20-float rows).
static __device__ __forceinline__ void tdm_load_tile_A(const float* gsrc, unsigned lds_off, int K)
{
    unsigned long long ga = (unsigned long long)gsrc;
    v4u g0;
    g0[0] = 1u;                                           // count=1 (valid user D#)
    g0[1] = lds_off;                                      // lds_addr
    g0[2] = (unsigned)(ga & 0xFFFFFFFFu);                 // global_addr[31:0]
    g0[3] = (unsigned)((ga >> 32) & 0x01FFFFFFu)          // global_addr[56:32]
          | (2u << 30);                                   // type=2 ("image")
    v8i g1;
    g1[0] = (2 << 16)        // data_size = 4B
          | (1 << 20)        // pad_enable
          | (3 << 22)        // pad_interval: 16 DWORDs
          | (3 << 25);       // pad_amount:   4 DWORDs
    g1[1] = (BK << 16);      // tensor_dim0 = 16   ([63:48])
    g1[2] = (BM << 16);      // tensor_dim1 = 128  ([95:80]); tensor_dim0[31:16]=0
    g1[3] = (BK << 16);      // tile_dim0  = 16    ([127:112]); tensor_dim1[31:16]=0
    g1[4] = BM;              // tile_dim1  = 128   ([143:128]); tile_dim2 = 0
    g1[5] = K;               // tensor_dim0_stride[31:0] (elements)
    g1[6] = 0;               // tensor_dim0_stride[47:32] = 0; tensor_dim1_stride lo = 0
    g1[7] = 0;
    v4i z4 = {0, 0, 0, 0};
    v8i z8 = {0, 0, 0, 0, 0, 0, 0, 0};
    __builtin_amdgcn_tensor_load_to_lds(g0, g1, z4, z4, z8, 0);
}
#endif

// MODE: 0 = bias only, 1 = bias + leaky_relu(0.2), 2 = residual: out = resid + v*0.1
template<int MODE>
__global__ __launch_bounds__(256)
void gemm_wmma_f32(const float* __restrict__ A, const float* __restrict__ W,
                   const float* __restrict__ bias, const float* __restrict__ resid,
                   float* __restrict__ out, int M, int N, int K)
{
    __shared__ float As [BM * LDA];   // [m][k] padded rows
    __shared__ float BsT[BN * BKP];   // [n][k] transposed -> contiguous (k,k+1) pairs

    const int tid    = threadIdx.x;
    const int lane   = tid & 31;
    const int wave   = tid >> 5;           // 0..7
    const int wm     = wave & 3;           // 32-row slab index
    const int wn     = wave >> 2;          // 64-col slab index
    const int lane16 = lane & 15;
    const int khalf  = (lane >> 4) << 1;   // 0 (lanes 0-15) or 2 (lanes 16-31)

    const int rowBase = blockIdx.y * BM;
    const int colBase = blockIdx.x * BN;

    v8f c[2][4];
    #pragma unroll
    for (int i = 0; i < 2; ++i)
        #pragma unroll
        for (int j = 0; j < 4; ++j)
            #pragma unroll
            for (int e = 0; e < 8; ++e)
                c[i][j][e] = 0.0f;

#if USE_TDM
    const unsigned asOff = lds_offset_of(As);
#endif

    for (int k0 = 0; k0 < K; k0 += BK) {
        // ---- stage A tile: BM x BK -> padded rows in LDS ----
#if USE_TDM
        if (wave == 0)   // uniform branch: exactly one wave issues the DMA
            tdm_load_tile_A(A + (size_t)rowBase * K + k0, asOff, K);
#else
        #pragma unroll
        for (int tci = 0; tci < 2; ++tci) {
            int chunk = tid + tci * 256;          // 512 float4 chunks
            int r  = chunk >> 2;                  // 0..127
            int c4 = (chunk & 3) << 2;            // 0,4,8,12
            v4f v = *(const v4f*)(A + (size_t)(rowBase + r) * K + (k0 + c4));
            *(v4f*)(As + r * LDA + c4) = v;
        }
#endif
        // ---- stage B tile: BK x BN, transposed into LDS ([n][k]) ----
        #pragma unroll
        for (int tci = 0; tci < 2; ++tci) {
            int chunk = tid + tci * 256;          // 512 float4 chunks
            int r  = chunk >> 5;                  // k row 0..15
            int c4 = (chunk & 31) << 2;           // n col 0..124
            v4f v = *(const v4f*)(W + (size_t)(k0 + r) * N + (colBase + c4));
            BsT[(c4 + 0) * BKP + r] = v.x;
            BsT[(c4 + 1) * BKP + r] = v.y;
            BsT[(c4 + 2) * BKP + r] = v.z;
            BsT[(c4 + 3) * BKP + r] = v.w;
        }
        // prefetch next W tile into the cache hierarchy (global_prefetch_b8)
        if (k0 + BK < K)
            __builtin_prefetch(W + (size_t)(k0 + BK + (tid >> 5)) * N + colBase + ((tid & 31) << 2), 0, 0);
#if USE_TDM
        __builtin_amdgcn_s_wait_tensorcnt(0);     // no-op for non-issuing waves
#endif
        __syncthreads();

        #pragma unroll
        for (int kk = 0; kk < BK; kk += 4) {
            v2f a[2];
            v2f b[4];
            #pragma unroll
            for (int i = 0; i < 2; ++i) {
                int m = wm * 32 + i * 16 + lane16;
                a[i] = *(const v2f*)(As + m * LDA + kk + khalf);        // A[m][k], A[m][k+1]
            }
            #pragma unroll
            for (int j = 0; j < 4; ++j) {
                int n = wn * 64 + j * 16 + lane16;
                b[j] = *(const v2f*)(BsT + n * BKP + kk + khalf);       // B[k][n], B[k+1][n]
            }
            #pragma unroll
            for (int i = 0; i < 2; ++i)
                #pragma unroll
                for (int j = 0; j < 4; ++j)
                    c[i][j] = __builtin_amdgcn_wmma_f32_16x16x4_f32(
                        false, a[i], false, b[j], (short)0, c[i][j], false, false);
        }
        __syncthreads();
    }

    // ---- fused epilogue: bias + activation / residual ----
    const int rhalf = (lane >> 4) << 3;   // C/D layout: lanes 16-31 hold M = r+8
    #pragma unroll
    for (int i = 0; i < 2; ++i) {
        #pragma unroll
        for (int j = 0; j < 4; ++j) {
            int col = colBase + wn * 64 + j * 16 + lane16;
            float bv = bias[col];
            #pragma unroll
            for (int r = 0; r < 8; ++r) {
                int row = rowBase + wm * 32 + i * 16 + r + rhalf;
                size_t o = (size_t)row * N + col;
                float v = c[i][j][r] + bv;
                if (MODE == 1) v = (v >= 0.0f) ? v : 0.2f * v;
                if (MODE == 2) v = resid[o] + v * 0.1f;
                out[o] = v;
            }
        }
    }
}

// csum[j] = sum_n coupling[j][n]
__global__ void csum_kernel(const float* __restrict__ coupling, float* __restrict__ csum)
{
    int j = threadIdx.x;
    if (j < 64) {
        float s = 0.0f;
        #pragma unroll 8
        for (int n = 0; n < 64; ++n) s += coupling[j * 64 + n];
        csum[j] = s;
    }
}

// Oscillator dynamics. sin(pj - pm) = sin(pj)cos(pm) - cos(pj)sin(pm) collapses
// the 64x64 einsum into two 64-wide reductions per token.
// 256 threads = 4 tokens x 64 oscillators.
__global__ __launch_bounds__(256)
void osc_dyn_kernel(const float* __restrict__ osc, const float* __restrict__ t,
                    const float* __restrict__ Wt, const float* __restrict__ bt,
                    const float* __restrict__ csum, const float* __restrict__ freqs,
                    const float* __restrict__ mu_p, float* __restrict__ comb, int S)
{
    __shared__ float rc[4][64];
    __shared__ float rs[4][64];

    const int g = threadIdx.x >> 6;        // token group within block (0..3)
    const int j = threadIdx.x & 63;        // oscillator index
    const int token = blockIdx.x * 4 + g;
    const int s = token % S;               // sequence position (batch boundary aware)

    const float w = csum[j];
    float pp = (s > 0) ? osc[(size_t)(token - 1) * 128 + j] : 0.0f;
    float sj = sinf(pp);
    float cj = cosf(pp);
    rc[g][j] = cj * w;
    rs[g][j] = sj * w;
    __syncthreads();

    float Sc = 0.0f, Ss = 0.0f;
    #pragma unroll 8
    for (int m = 0; m < 64; ++m) { Sc += rc[g][m]; Ss += rs[g][m]; }

    const size_t o = (size_t)token * 128;
    if (s == 0) {
        comb[o + j]      = osc[o + j];
        comb[o + 64 + j] = osc[o + 64 + j];
    } else {
        float coupling_sum = sj * Sc - cj * Ss;
        float tmod   = t[token] * Wt[j] + bt[j];
        float dphase = freqs[j] + coupling_sum * 0.1f + tmod * 0.1f;
        comb[o + j]  = pp + dphase * 0.1f;

        float ap   = osc[(size_t)(token - 1) * 128 + 64 + j];
        float mu   = mu_p[0];
        float damp = (-0.1f + mu) * ap - ap * ap * ap;
        comb[o + 64 + j] = tanhf(ap + damp * 0.1f) * 1.0f;   // AMP_SAT = 1.0
    }
}

extern "C" void kernel_launch(void* const* d_in, const int* in_sizes, int n_in,
                              void* d_out, int out_size, void* d_ws, size_t ws_size,
                              hipStream_t stream)
{
    const float* x    = (const float*)d_in[0];
    const float* t    = (const float*)d_in[1];
    const float* W1   = (const float*)d_in[2];
    const float* b1   = (const float*)d_in[3];
    const float* W2   = (const float*)d_in[4];
    const float* b2   = (const float*)d_in[5];
    const float* W3   = (const float*)d_in[6];
    const float* b3   = (const float*)d_in[7];
    const float* W4   = (const float*)d_in[8];
    const float* b4   = (const float*)d_in[9];
    const float* Wt   = (const float*)d_in[10];
    const float* bt   = (const float*)d_in[11];
    const float* coup = (const float*)d_in[12];
    const float* freq = (const float*)d_in[13];
    const float* mu   = (const float*)d_in[14];
    float* out = (float*)d_out;

    const int S = 2048;
    const int Mtok = 4 * S;        // 8192
    const int H = 1024;
    const int NOSC2 = 128;

    // workspace layout (fp32): h/h2 (reused) | osc | combined | csum
    float* h    = (float*)d_ws;
    float* oscb = h    + (size_t)Mtok * H;
    float* comb = oscb + (size_t)Mtok * NOSC2;
    float* csum = comb + (size_t)Mtok * NOSC2;

    csum_kernel<<<1, 64, 0, stream>>>(coup, csum);

    // h = leaky_relu(x @ W1 + b1)
    gemm_wmma_f32<1><<<dim3(H / BN, Mtok / BM), 256, 0, stream>>>(
        x, W1, b1, nullptr, h, Mtok, H, H);

    // osc = h @ W2 + b2
    gemm_wmma_f32<0><<<dim3(NOSC2 / BN, Mtok / BM), 256, 0, stream>>>(
        h, W2, b2, nullptr, oscb, Mtok, NOSC2, H);

    // combined = oscillator dynamics (phase_out || amp_out)
    osc_dyn_kernel<<<Mtok / 4, 256, 0, stream>>>(
        oscb, t, Wt, bt, csum, freq, mu, comb, S);

    // h2 = leaky_relu(combined @ W3 + b3)   (reuses h buffer)
    gemm_wmma_f32<1><<<dim3(H / BN, Mtok / BM), 256, 0, stream>>>(
        comb, W3, b3, nullptr, h, Mtok, H, NOSC2);

    // out = x + (h2 @ W4 + b4) * 0.1
    gemm_wmma_f32<2><<<dim3(H / BN, Mtok / BM), 256, 0, stream>>>(
        h, W4, b4, x, out, Mtok, H, H);
}